// HGNN_DB_45749991637711
// MI455X (gfx1250) — compile-verified
//
#include <hip/hip_runtime.h>
#include <hip/hip_bf16.h>

#define N_NODES 50000
#define E_EDGES 1600000
#define IN_DIM  256
#define D_DIM   64
#define P_META  2
#define K_STEPS 3
#define ALPHA_  0.1f
#define BETA_   0.5f
#define GAMMA_  0.5f

typedef float v2f __attribute__((ext_vector_type(2)));
typedef float v8f __attribute__((ext_vector_type(8)));

__device__ __forceinline__ v8f wmma4(v2f a, v2f b, v8f c) {
  // V_WMMA_F32_16X16X4_F32 : full-fp32 matrix op, 8-arg VOP3P form
  return __builtin_amdgcn_wmma_f32_16x16x4_f32(
      false, a, false, b, (short)0, c, false, false);
}

// ---------------- degree histogram ----------------
__global__ void k_deg(const int* __restrict__ src, const int* __restrict__ dst,
                      float* __restrict__ deg_s, float* __restrict__ deg_d) {
  int t = blockIdx.x * blockDim.x + threadIdx.x;
  if (t >= P_META * E_EDGES) return;
  int p = t / E_EDGES;
  atomicAdd(deg_s + (size_t)p * N_NODES + src[t], 1.0f);
  atomicAdd(deg_d + (size_t)p * N_NODES + dst[t], 1.0f);
}

// ---------------- per-node normalizations ----------------
__global__ void k_norms(const float* __restrict__ deg_s, const float* __restrict__ deg_d,
                        float* __restrict__ ns, float* __restrict__ nd,
                        float* __restrict__ invdd) {
  int i = blockIdx.x * blockDim.x + threadIdx.x;
  if (i >= P_META * N_NODES) return;
  float ds = deg_s[i]; if (ds < 1.f) ds = 1.f;
  float dd = deg_d[i]; if (dd < 1.f) dd = 1.f;
  ns[i] = rsqrtf(ds);
  nd[i] = rsqrtf(dd);
  invdd[i] = 1.f / dd;
}

// ---------------- feature projection: H = elu(feats @ fcs_w + b) ----------------
__global__ __launch_bounds__(128) void k_proj(const float* __restrict__ A,
                                              const float* __restrict__ W,
                                              const float* __restrict__ bias,
                                              float* __restrict__ H) {
  const int m0   = blockIdx.x * 16;
  const int wave = threadIdx.x >> 5;
  const int lane = threadIdx.x & 31;
  const int half = lane >> 4;
  const int r    = lane & 15;
  const int n0   = wave * 16;
  const int col  = n0 + r;
  v8f acc = {};
  const float* arow = A + (size_t)(m0 + r) * IN_DIM + 2 * half;
  for (int k0 = 0; k0 < IN_DIM; k0 += 4) {
    v2f a, b;
    a.x = arow[k0];
    a.y = arow[k0 + 1];
    b.x = W[(size_t)(k0 + 2 * half)     * D_DIM + col];
    b.y = W[(size_t)(k0 + 2 * half + 1) * D_DIM + col];
    acc = wmma4(a, b, acc);
  }
  float bv = bias[col];
#pragma unroll
  for (int i = 0; i < 8; ++i) {
    int row = m0 + i + 8 * half;
    float x = acc[i] + bv;
    H[(size_t)row * D_DIM + col] = x > 0.f ? x : (__expf(x) - 1.f);
  }
}

// ---------------- edge scatter: out[dst] += X[src] * (scale ? scale[src] : 1) ----
__global__ void k_scatter(const int* __restrict__ src, const int* __restrict__ dst,
                          const float* __restrict__ X, const float* __restrict__ scale,
                          float* __restrict__ out) {
  int t = blockIdx.x * blockDim.x + threadIdx.x;
  int e = t >> 4;
  if (e >= E_EDGES) return;
  int c = (t & 15) << 2;
  int s = src[e], d = dst[e];
  float sc = scale ? scale[s] : 1.0f;
  const float4 v = *(const float4*)(X + (size_t)s * D_DIM + c);
  float* o = out + (size_t)d * D_DIM + c;
  atomicAdd(o + 0, v.x * sc);
  atomicAdd(o + 1, v.y * sc);
  atomicAdd(o + 2, v.z * sc);
  atomicAdd(o + 3, v.w * sc);
}

// ---------------- conv: Z = prelu((S/deg_dst) @ W + b) ----------------
__global__ __launch_bounds__(128) void k_conv(const float* __restrict__ S,
                                              const float* __restrict__ invdd_p,
                                              const float* __restrict__ W,
                                              const float* __restrict__ bvec,
                                              const float* __restrict__ prelu_p,
                                              float* __restrict__ Zp) {
  const int m0   = blockIdx.x * 16;
  const int wave = threadIdx.x >> 5;
  const int lane = threadIdx.x & 31;
  const int half = lane >> 4;
  const int r    = lane & 15;
  const int n0   = wave * 16;
  const int col  = n0 + r;
  const float inv = invdd_p[m0 + r];
  v8f acc = {};
  const float* arow = S + (size_t)(m0 + r) * D_DIM + 2 * half;
  for (int k0 = 0; k0 < D_DIM; k0 += 4) {
    v2f a, b;
    a.x = arow[k0]     * inv;
    a.y = arow[k0 + 1] * inv;
    b.x = W[(size_t)(k0 + 2 * half)     * D_DIM + col];
    b.y = W[(size_t)(k0 + 2 * half + 1) * D_DIM + col];
    acc = wmma4(a, b, acc);
  }
  float bv = bvec[col];
  float slope = *prelu_p;
#pragma unroll
  for (int i = 0; i < 8; ++i) {
    int row = m0 + i + 8 * half;
    float g = acc[i] + bv;
    Zp[(size_t)row * D_DIM + col] = g > 0.f ? g : slope * g;
  }
}

// ---------------- APPNP mid (in place): S = (1-a)*S*nd + a*H0 ----------------
__global__ void k_appnp_mid(float* __restrict__ S, const float* __restrict__ nd_p,
                            const float* __restrict__ H0) {
  int i = blockIdx.x * blockDim.x + threadIdx.x;
  if (i >= N_NODES * D_DIM) return;
  int row = i >> 6;
  S[i] = (1.f - ALPHA_) * S[i] * nd_p[row] + ALPHA_ * H0[i];
}

// ---------------- APPNP gemm: Out = (1-b)*M + b*(M @ W) ----------------
__global__ __launch_bounds__(128) void k_appnp_gemm(const float* __restrict__ M,
                                                    const float* __restrict__ W,
                                                    float* __restrict__ Out) {
  const int m0   = blockIdx.x * 16;
  const int wave = threadIdx.x >> 5;
  const int lane = threadIdx.x & 31;
  const int half = lane >> 4;
  const int r    = lane & 15;
  const int n0   = wave * 16;
  const int col  = n0 + r;
  v8f acc = {};
  const float* arow = M + (size_t)(m0 + r) * D_DIM + 2 * half;
  for (int k0 = 0; k0 < D_DIM; k0 += 4) {
    v2f a, b;
    a.x = arow[k0];
    a.y = arow[k0 + 1];
    b.x = W[(size_t)(k0 + 2 * half)     * D_DIM + col];
    b.y = W[(size_t)(k0 + 2 * half + 1) * D_DIM + col];
    acc = wmma4(a, b, acc);
  }
#pragma unroll
  for (int i = 0; i < 8; ++i) {
    int row = m0 + i + 8 * half;
    float m = M[(size_t)row * D_DIM + col];
    Out[(size_t)row * D_DIM + col] = (1.f - BETA_) * m + BETA_ * acc[i];
  }
}

// ---------------- attention partial: wsum[p][col] += sum_rows tanh(Z@W + b) ------
__global__ __launch_bounds__(128) void k_attn(const float* __restrict__ Z,
                                              const float* __restrict__ W,
                                              const float* __restrict__ bvec,
                                              float* __restrict__ wsum) {
  const int p    = blockIdx.y;
  const float* A = Z + (size_t)p * N_NODES * D_DIM;
  const int m0   = blockIdx.x * 16;
  const int wave = threadIdx.x >> 5;
  const int lane = threadIdx.x & 31;
  const int half = lane >> 4;
  const int r    = lane & 15;
  const int n0   = wave * 16;
  const int col  = n0 + r;
  v8f acc = {};
  const float* arow = A + (size_t)(m0 + r) * D_DIM + 2 * half;
  for (int k0 = 0; k0 < D_DIM; k0 += 4) {
    v2f a, b;
    a.x = arow[k0];
    a.y = arow[k0 + 1];
    b.x = W[(size_t)(k0 + 2 * half)     * D_DIM + col];
    b.y = W[(size_t)(k0 + 2 * half + 1) * D_DIM + col];
    acc = wmma4(a, b, acc);
  }
  float bv = bvec[col];
  float sum = 0.f;
#pragma unroll
  for (int i = 0; i < 8; ++i) sum += tanhf(acc[i] + bv);
  sum += __shfl_xor(sum, 16, 32);           // fold the two row-halves per column
  if (half == 0) atomicAdd(&wsum[p * D_DIM + col], sum);
}

// ---------------- softmax over P=2 semantic scores ----------------
__global__ void k_beta(const float* __restrict__ wsum1, const float* __restrict__ wsum2,
                       const float* __restrict__ q, float* __restrict__ betas) {
  if (threadIdx.x != 0 || blockIdx.x != 0) return;
  const float invN = 1.0f / (float)N_NODES;
  float sc1[P_META], sc2[P_META];
  for (int p = 0; p < P_META; ++p) {
    float s1 = 0.f, s2 = 0.f;
    for (int d = 0; d < D_DIM; ++d) {
      s1 += wsum1[p * D_DIM + d] * invN * q[d];
      s2 += wsum2[p * D_DIM + d] * invN * q[d];
    }
    sc1[p] = s1; sc2[p] = s2;
  }
  float m1 = fmaxf(sc1[0], sc1[1]);
  float e10 = __expf(sc1[0] - m1), e11 = __expf(sc1[1] - m1);
  betas[0] = e10 / (e10 + e11); betas[1] = e11 / (e10 + e11);
  float m2 = fmaxf(sc2[0], sc2[1]);
  float e20 = __expf(sc2[0] - m2), e21 = __expf(sc2[1] - m2);
  betas[2] = e20 / (e20 + e21); betas[3] = e21 / (e20 + e21);
}

// ---------------- final combine ----------------
__global__ void k_combine(const float* __restrict__ z1, const float* __restrict__ z2,
                          const float* __restrict__ betas, float* __restrict__ out) {
  int i = blockIdx.x * blockDim.x + threadIdx.x;
  if (i >= N_NODES * D_DIM) return;
  size_t off = (size_t)N_NODES * D_DIM;
  out[i] = GAMMA_ * (betas[0] * z1[i] + betas[1] * z1[i + off]) +
           (1.f - GAMMA_) * (betas[2] * z2[i] + betas[3] * z2[i + off]);
}

extern "C" void kernel_launch(void* const* d_in, const int* in_sizes, int n_in,
                              void* d_out, int out_size, void* d_ws, size_t ws_size,
                              hipStream_t stream) {
  const float* feats      = (const float*)d_in[0];
  const float* fcs_w      = (const float*)d_in[1];
  const float* fcs_b      = (const float*)d_in[2];
  const float* conv_w     = (const float*)d_in[3];
  const float* conv_b     = (const float*)d_in[4];
  const float* conv_prelu = (const float*)d_in[5];
  const float* appnp_w    = (const float*)d_in[6];
  const float* attn_w     = (const float*)d_in[7];
  const float* attn_b     = (const float*)d_in[8];
  const float* attn_q     = (const float*)d_in[9];
  const int*   src        = (const int*)d_in[10];
  const int*   dst        = (const int*)d_in[11];
  float*       out        = (float*)d_out;

  const size_t ND = (size_t)N_NODES * D_DIM;
  float* ws    = (float*)d_ws;
  float* h     = ws;  ws += ND;
  float* S     = ws;  ws += ND;
  float* hpA   = ws;  ws += ND;
  float* hpB   = ws;  ws += ND;
  float* z1    = ws;  ws += (size_t)P_META * ND;
  float* z2    = ws;  ws += (size_t)P_META * ND;
  float* deg_s = ws;  ws += (size_t)P_META * N_NODES;
  float* deg_d = ws;  ws += (size_t)P_META * N_NODES;
  float* nsArr = ws;  ws += (size_t)P_META * N_NODES;
  float* ndArr = ws;  ws += (size_t)P_META * N_NODES;
  float* invdd = ws;  ws += (size_t)P_META * N_NODES;
  float* wsum1 = ws;  ws += P_META * D_DIM;
  float* wsum2 = ws;  ws += P_META * D_DIM;
  float* betas = ws;  ws += 4;

  const int MT = N_NODES / 16;  // 3125 row tiles

  // degrees + norms
  hipMemsetAsync(deg_s, 0, 2 * (size_t)P_META * N_NODES * sizeof(float), stream);
  k_deg<<<(P_META * E_EDGES + 255) / 256, 256, 0, stream>>>(src, dst, deg_s, deg_d);
  k_norms<<<(P_META * N_NODES + 255) / 256, 256, 0, stream>>>(deg_s, deg_d, nsArr, ndArr, invdd);

  // projection
  k_proj<<<MT, 128, 0, stream>>>(feats, fcs_w, fcs_b, h);

  for (int p = 0; p < P_META; ++p) {
    const int* sp = src + (size_t)p * E_EDGES;
    const int* dp = dst + (size_t)p * E_EDGES;

    // GraphConv branch -> z1[p]
    hipMemsetAsync(S, 0, ND * sizeof(float), stream);
    k_scatter<<<(E_EDGES * 16 + 255) / 256, 256, 0, stream>>>(sp, dp, h, nullptr, S);
    k_conv<<<MT, 128, 0, stream>>>(S, invdd + (size_t)p * N_NODES,
                                   conv_w + (size_t)p * D_DIM * D_DIM,
                                   conv_b + (size_t)p * D_DIM,
                                   conv_prelu + p,
                                   z1 + (size_t)p * ND);

    // APPNP branch -> z2[p]
    const float* hin = h;
    float* houts[K_STEPS] = { hpA, hpB, z2 + (size_t)p * ND };
    for (int k = 0; k < K_STEPS; ++k) {
      hipMemsetAsync(S, 0, ND * sizeof(float), stream);
      k_scatter<<<(E_EDGES * 16 + 255) / 256, 256, 0, stream>>>(
          sp, dp, hin, nsArr + (size_t)p * N_NODES, S);
      k_appnp_mid<<<(int)((ND + 255) / 256), 256, 0, stream>>>(
          S, ndArr + (size_t)p * N_NODES, h);
      k_appnp_gemm<<<MT, 128, 0, stream>>>(S, appnp_w + (size_t)p * D_DIM * D_DIM, houts[k]);
      hin = houts[k];
    }
  }

  // semantic attention
  hipMemsetAsync(wsum1, 0, 2 * P_META * D_DIM * sizeof(float), stream);
  k_attn<<<dim3(MT, P_META), 128, 0, stream>>>(z1, attn_w, attn_b, wsum1);
  k_attn<<<dim3(MT, P_META), 128, 0, stream>>>(z2, attn_w, attn_b, wsum2);
  k_beta<<<1, 1, 0, stream>>>(wsum1, wsum2, attn_q, betas);
  k_combine<<<(int)((ND + 255) / 256), 256, 0, stream>>>(z1, z2, betas, out);
}